// MolConv_9517647528175
// MI455X (gfx1250) — compile-verified
//
#include <hip/hip_runtime.h>
#include <math.h>

// Problem constants (match reference)
#define Bd   32
#define Dd   21
#define Nn   1024
#define Kk   32
#define OUTC 64
#define DP   32     // padded feature dim for xtp rows (zero-filled 21..31)
#define NB12 1024   // blocks for scalar-stat partials
#define NBUP 8192   // blocks for up-stat partials (= B*N*K/128)

typedef __attribute__((ext_vector_type(2))) float v2f;
typedef __attribute__((ext_vector_type(8))) float v8f;

__device__ __forceinline__ v8f wmma_f32_16x16x4(v2f a, v2f b, v8f c) {
  // (neg_a, A, neg_b, B, c_mod, C, reuse_a, reuse_b)
  return __builtin_amdgcn_wmma_f32_16x16x4_f32(false, a, false, b, (short)0, c,
                                               false, false);
}

__device__ __forceinline__ float sigm(float x) { return 1.f / (1.f + expf(-x)); }

// ---------------------------------------------------------------------------
// K0: xtp[b][n][0..31] = x[b][:,n] (zero padded), xx[b][n] = sum_d x^2
// ---------------------------------------------------------------------------
__global__ __launch_bounds__(256) void prep_kernel(
    const float* __restrict__ x, float* __restrict__ xtp, float* __restrict__ xx) {
  int t = blockIdx.x * 256 + threadIdx.x;           // over B*N
  if (t >= Bd * Nn) return;
  int b = t / Nn, n = t % Nn;
  const float* xp = x + (size_t)b * Dd * Nn + n;
  float v[DP];
#pragma unroll
  for (int d = 0; d < DP; ++d) v[d] = 0.f;
  float s = 0.f;
#pragma unroll
  for (int d = 0; d < Dd; ++d) {
    float q = xp[(size_t)d * Nn];
    v[d] = q;
    s += q * q;
  }
  float* o = xtp + (size_t)t * DP;
#pragma unroll
  for (int d = 0; d < DP; ++d) o[d] = v[d];
  xx[t] = s;
}

// ---------------------------------------------------------------------------
// K1: pd[m][n] = 2*<x_m,x_n> - |x_m|^2 - |x_n|^2 via WMMA, then top-32 per row.
// One block per (b, 16-row strip). pd strip (16 x 1024) lives in LDS.
// ---------------------------------------------------------------------------
__global__ __launch_bounds__(256) void knn_kernel(
    const float* __restrict__ xtp, const float* __restrict__ xx,
    float* __restrict__ dist, int* __restrict__ idxo) {
  __shared__ float pd[16][Nn];                       // 64 KB
  int b = blockIdx.x >> 6;                           // N/16 = 64 strips per b
  int m0 = (blockIdx.x & 63) << 4;
  int wave = threadIdx.x >> 5;
  int lane = threadIdx.x & 31;
  int lo = lane & 15, hi = lane >> 4;

  const float* xb = xtp + (size_t)b * Nn * DP;
  const float* xxb = xx + (size_t)b * Nn;

  // A fragments for the 16 center rows, all 6 k-steps (K-dim = padded 24)
  v2f afrag[6];
#pragma unroll
  for (int i = 0; i < 6; ++i)
    afrag[i] = *(const v2f*)(xb + (size_t)(m0 + lo) * DP + (4 * i + 2 * hi));
  float xxr[8];
#pragma unroll
  for (int v = 0; v < 8; ++v) xxr[v] = xxb[m0 + v + 8 * hi];

  for (int ct = wave; ct < 64; ct += 8) {
    int n0 = ct << 4;
    if (ct + 8 < 64)
      __builtin_prefetch(xb + (size_t)((ct + 8) * 16 + lo) * DP, 0, 1);
    v8f acc = {};
#pragma unroll
    for (int i = 0; i < 6; ++i) {
      v2f bfrag = *(const v2f*)(xb + (size_t)(n0 + lo) * DP + (4 * i + 2 * hi));
      acc = wmma_f32_16x16x4(afrag[i], bfrag, acc);
    }
    float xxc = xxb[n0 + lo];
#pragma unroll
    for (int v = 0; v < 8; ++v)
      pd[v + 8 * hi][n0 + lo] = 2.f * acc[v] - xxr[v] - xxc;
  }
  __syncthreads();

  // top-32 per row; 2 rows per wave; tie-break: smaller column wins.
  for (int rr = 0; rr < 2; ++rr) {
    int r = 2 * wave + rr;
    int n = m0 + r;
    float vals[32];
#pragma unroll
    for (int j = 0; j < 32; ++j) vals[j] = pd[r][lane + 32 * j];
    for (int k = 0; k < Kk; ++k) {
      float bv = -3.402823e38f;
      int bc = 0x7fffffff;
#pragma unroll
      for (int j = 0; j < 32; ++j) {
        int c = lane + 32 * j;
        if (vals[j] > bv || (vals[j] == bv && c < bc)) { bv = vals[j]; bc = c; }
      }
#pragma unroll
      for (int s = 16; s >= 1; s >>= 1) {
        float ov = __shfl_xor(bv, s, 32);
        int oc = __shfl_xor(bc, s, 32);
        if (ov > bv || (ov == bv && oc < bc)) { bv = ov; bc = oc; }
      }
      if ((bc & 31) == lane) {
#pragma unroll
        for (int j = 0; j < 32; ++j)
          if (j == (bc >> 5)) vals[j] = -3.402823e38f;
      }
      if (lane == k) {
        dist[((size_t)b * Nn + n) * Kk + k] = -bv;
        idxo[((size_t)b * Nn + n) * Kk + k] = bc;
      }
    }
  }
}

// ---------------------------------------------------------------------------
// K2: sub[b,n,k] = <x_{idx[b,n,k]}, x_{idx[b,n,0]}>
// ---------------------------------------------------------------------------
__global__ __launch_bounds__(256) void sub_kernel(
    const float* __restrict__ xtp, const int* __restrict__ idxo,
    float* __restrict__ sub) {
  size_t t = (size_t)blockIdx.x * 256 + threadIdx.x;  // over B*N*K
  if (t >= (size_t)Bd * Nn * Kk) return;
  size_t bn = t / Kk;
  int b = (int)(bn / Nn);
  const float* xb = xtp + (size_t)b * Nn * DP;
  const float* p0 = xb + (size_t)idxo[bn * Kk] * DP;
  const float* pk = xb + (size_t)idxo[t] * DP;
  float s = 0.f;
#pragma unroll
  for (int d = 0; d < Dd; ++d) s += p0[d] * pk[d];
  sub[t] = s;
}

// ---------------------------------------------------------------------------
// K3: S[b,k,j] = sum_n sub^2[b,n,k]*sub^2[b,n,j]  (WMMA, K-dim = N = 1024)
//     Mmat[b,k,j] = w_gm[j] / max(sqrt(S), 1e-12)
// ---------------------------------------------------------------------------
__global__ __launch_bounds__(128) void normgram_kernel(
    const float* __restrict__ sub, const float* __restrict__ wgm,
    float* __restrict__ Mmat) {
  int b = blockIdx.x;
  int wave = threadIdx.x >> 5, lane = threadIdx.x & 31;
  int lo = lane & 15, hi = lane >> 4;
  int k0 = (wave >> 1) << 4, j0 = (wave & 1) << 4;
  const float* sb = sub + (size_t)b * Nn * Kk;
  v8f acc = {};
  for (int n0 = 0; n0 < Nn; n0 += 4) {
    int ra = n0 + 2 * hi;
    float ax = sb[(size_t)ra * Kk + k0 + lo];
    float ay = sb[(size_t)(ra + 1) * Kk + k0 + lo];
    float bx = sb[(size_t)ra * Kk + j0 + lo];
    float by = sb[(size_t)(ra + 1) * Kk + j0 + lo];
    v2f a = {ax * ax, ay * ay};
    v2f bb = {bx * bx, by * by};
    acc = wmma_f32_16x16x4(a, bb, acc);
  }
#pragma unroll
  for (int v = 0; v < 8; ++v) {
    int kk = k0 + v + 8 * hi, jj = j0 + lo;
    float nrm = sqrtf(acc[v]);
    Mmat[((size_t)b * Kk + kk) * Kk + jj] = wgm[jj] / fmaxf(nrm, 1e-12f);
  }
}

// ---------------------------------------------------------------------------
// K4: G[b,n,k] = sum_j sub[b,n,j]*Mmat[b,k,j];  gmred = sub .* G  (WMMA)
// ---------------------------------------------------------------------------
__global__ __launch_bounds__(256) void gmred_kernel(
    const float* __restrict__ sub, const float* __restrict__ Mmat,
    float* __restrict__ gmred) {
  int tile = blockIdx.x * 8 + (threadIdx.x >> 5);    // 4096 tiles total
  int lane = threadIdx.x & 31, lo = lane & 15, hi = lane >> 4;
  int b = tile >> 7;                                 // 128 tiles per batch
  int rem = tile & 127;
  int n0 = (rem >> 1) << 4;
  int k0 = (rem & 1) << 4;
  const float* sb = sub + (size_t)b * Nn * Kk;
  const float* Mb = Mmat + (size_t)b * Kk * Kk;
  v8f acc = {};
#pragma unroll
  for (int j0 = 0; j0 < Kk; j0 += 4) {
    v2f a = *(const v2f*)(sb + (size_t)(n0 + lo) * Kk + j0 + 2 * hi);
    v2f bb = *(const v2f*)(Mb + (size_t)(k0 + lo) * Kk + j0 + 2 * hi);
    acc = wmma_f32_16x16x4(a, bb, acc);
  }
#pragma unroll
  for (int v = 0; v < 8; ++v) {
    int nn = n0 + v + 8 * hi, kk = k0 + lo;
    size_t o = ((size_t)b * Nn + nn) * Kk + kk;
    gmred[o] = sb[(size_t)nn * Kk + kk] * acc[v];
  }
}

// ---------------------------------------------------------------------------
// K5: per-block partial sums for BN1 (dist*w_dist) and BN2 (gmred)
// ---------------------------------------------------------------------------
__global__ __launch_bounds__(256) void stats12_kernel(
    const float* __restrict__ dist, const float* __restrict__ gmred,
    const float* __restrict__ wdist, float* __restrict__ part) {
  __shared__ float4 red[256];
  float wd = wdist[0];
  size_t total = (size_t)Bd * Nn * Kk;
  float s1 = 0, q1 = 0, s2 = 0, q2 = 0;
  for (size_t i = (size_t)blockIdx.x * 256 + threadIdx.x; i < total;
       i += (size_t)NB12 * 256) {
    float d1 = dist[i] * wd;
    s1 += d1; q1 += d1 * d1;
    float g = gmred[i];
    s2 += g; q2 += g * g;
  }
  red[threadIdx.x] = make_float4(s1, q1, s2, q2);
  __syncthreads();
  for (int s = 128; s >= 1; s >>= 1) {
    if ((int)threadIdx.x < s) {
      float4 a = red[threadIdx.x], c = red[threadIdx.x + s];
      red[threadIdx.x] = make_float4(a.x + c.x, a.y + c.y, a.z + c.z, a.w + c.w);
    }
    __syncthreads();
  }
  if (threadIdx.x == 0) {
    float4 r = red[0];
    part[(size_t)blockIdx.x * 4 + 0] = r.x;
    part[(size_t)blockIdx.x * 4 + 1] = r.y;
    part[(size_t)blockIdx.x * 4 + 2] = r.z;
    part[(size_t)blockIdx.x * 4 + 3] = r.w;
  }
}

__global__ void reduce12_kernel(const float* __restrict__ part,
                                float* __restrict__ stats12) {
  int t = threadIdx.x;
  if (t < 4) {
    float s = 0.f;
    for (int i = 0; i < NB12; ++i) s += part[(size_t)i * 4 + t];
    stats12[t] = s;
  }
}

// ---------------------------------------------------------------------------
// K6/K7: gate + feat (LDS) + up = Wup*feat (WMMA).
// STATS=1: write per-block per-channel sum/sumsq partials.
// STATS=0: BN3 + leaky + mean over K -> out[b][o][n].
// ---------------------------------------------------------------------------
template <int STATS>
__global__ __launch_bounds__(256) void up_kernel(
    const float* __restrict__ xtp, const float* __restrict__ dist,
    const int* __restrict__ idxo, const float* __restrict__ gmred,
    const float* __restrict__ stats12, const float* __restrict__ wdist,
    const float* __restrict__ g1, const float* __restrict__ b1,
    const float* __restrict__ g2, const float* __restrict__ b2,
    const float* __restrict__ wup, const float* __restrict__ g3,
    const float* __restrict__ b3, const float* __restrict__ statsup,
    float* __restrict__ partup, float* __restrict__ out) {
  __shared__ float featL[24][128];   // 12 KB, feat tile (d-major)
  __shared__ float WL[64][24];       //  6 KB, padded Wup
  __shared__ float upL[64][128];     // 32 KB, up results
  __shared__ float ps[4][64], pq[4][64];

  int lane = threadIdx.x & 31, wave = threadIdx.x >> 5;
  int lo = lane & 15, hi = lane >> 4;

  for (int i = threadIdx.x; i < 64 * 24; i += 256) {
    int o = i / 24, d = i % 24;
    WL[o][d] = (d < Dd) ? wup[o * Dd + d] : 0.f;
  }

  const float cnt = (float)((size_t)Bd * Nn * Kk);
  float mu1 = stats12[0] / cnt, var1 = stats12[1] / cnt - mu1 * mu1;
  float inv1 = rsqrtf(var1 + 1e-5f);
  float mu2 = stats12[2] / cnt, var2 = stats12[3] / cnt - mu2 * mu2;
  float inv2 = rsqrtf(var2 + 1e-5f);

  size_t col0 = (size_t)blockIdx.x * 128;
  if (threadIdx.x < 128) {
    size_t cg = col0 + threadIdx.x;
    int b = (int)(cg / ((size_t)Nn * Kk));
    int n = (int)((cg / Kk) % Nn);
    float d1 = dist[cg] * wdist[0];
    float w1 = sigm((d1 - mu1) * inv1 * g1[0] + b1[0]);
    float w2 = sigm((gmred[cg] - mu2) * inv2 * g2[0] + b2[0]);
    float gate = w1 * w2;
    const float* fn = xtp + ((size_t)b * Nn * Kk / Kk + idxo[cg]) * DP
                      + (size_t)b * 0;  // neighbor row in batch b
    fn = xtp + ((size_t)b * Nn + idxo[cg]) * DP;
    const float* fc = xtp + ((size_t)b * Nn + n) * DP;
#pragma unroll
    for (int d = 0; d < Dd; ++d)
      featL[d][threadIdx.x] = gate * fn[d] + (1.f - gate) * fc[d];
#pragma unroll
    for (int d = Dd; d < 24; ++d) featL[d][threadIdx.x] = 0.f;
  }
  __syncthreads();

  // 64x128x24 GEMM: 4 o-tiles x 8 col-tiles = 32 tiles, 4 per wave
#pragma unroll
  for (int tt = 0; tt < 4; ++tt) {
    int tile = wave * 4 + tt;
    int o0 = (tile >> 3) << 4, c0 = (tile & 7) << 4;
    v8f acc = {};
#pragma unroll
    for (int i = 0; i < 6; ++i) {
      int d0 = 4 * i + 2 * hi;
      v2f a = {WL[o0 + lo][d0], WL[o0 + lo][d0 + 1]};
      v2f bb = {featL[d0][c0 + lo], featL[d0 + 1][c0 + lo]};
      acc = wmma_f32_16x16x4(a, bb, acc);
    }
#pragma unroll
    for (int v = 0; v < 8; ++v) upL[o0 + v + 8 * hi][c0 + lo] = acc[v];
  }
  __syncthreads();

  int o = threadIdx.x & 63, q = threadIdx.x >> 6;   // q = 0..3 (column group)
  if (STATS) {
    float s = 0.f, sq = 0.f;
#pragma unroll
    for (int c = 0; c < 32; ++c) {
      float u = upL[o][q * 32 + c];
      s += u; sq += u * u;
    }
    ps[q][o] = s; pq[q][o] = sq;
    __syncthreads();
    if (threadIdx.x < 64) {
      int oo = threadIdx.x;
      partup[(size_t)blockIdx.x * 128 + oo] =
          ps[0][oo] + ps[1][oo] + ps[2][oo] + ps[3][oo];
      partup[(size_t)blockIdx.x * 128 + 64 + oo] =
          pq[0][oo] + pq[1][oo] + pq[2][oo] + pq[3][oo];
    }
  } else {
    float mu = statsup[o] / cnt;
    float var = statsup[64 + o] / cnt - mu * mu;
    float inv = rsqrtf(var + 1e-5f);
    float sc = inv * g3[o];
    float of = b3[o] - mu * sc;
    float s = 0.f;
#pragma unroll
    for (int c = 0; c < 32; ++c) {
      float u = upL[o][q * 32 + c] * sc + of;
      u = (u >= 0.f) ? u : 0.02f * u;
      s += u;
    }
    size_t cg = col0 + (size_t)q * 32;
    int b = (int)(cg / ((size_t)Nn * Kk));
    int n = (int)((cg / Kk) % Nn);
    out[((size_t)b * OUTC + o) * Nn + n] = s * (1.f / 32.f);
  }
}

__global__ void reduceup_kernel(const float* __restrict__ partup,
                                float* __restrict__ statsup) {
  int t = threadIdx.x;  // 128 threads: 64 sums + 64 sumsq
  float s = 0.f;
  for (int blk = 0; blk < NBUP; ++blk) s += partup[(size_t)blk * 128 + t];
  statsup[t] = s;
}

// ---------------------------------------------------------------------------
extern "C" void kernel_launch(void* const* d_in, const int* in_sizes, int n_in,
                              void* d_out, int out_size, void* d_ws,
                              size_t ws_size, hipStream_t stream) {
  const float* x      = (const float*)d_in[0];   // (B,D,N)
  // d_in[1] = idx_base (unused; gather offset computed from b directly)
  const float* w_dist = (const float*)d_in[2];
  const float* g1     = (const float*)d_in[3];
  const float* b1     = (const float*)d_in[4];
  const float* w_gm   = (const float*)d_in[5];   // (1,K)
  const float* g2     = (const float*)d_in[6];
  const float* b2     = (const float*)d_in[7];
  const float* w_up   = (const float*)d_in[8];   // (OUT,D)
  const float* g3     = (const float*)d_in[9];
  const float* b3     = (const float*)d_in[10];
  float* out = (float*)d_out;                    // (B,OUT,N)

  float* base = (float*)d_ws;
  size_t off = 0;
  float* xtp    = base + off; off += (size_t)Bd * Nn * DP;
  float* xx     = base + off; off += (size_t)Bd * Nn;
  float* dist   = base + off; off += (size_t)Bd * Nn * Kk;
  int*   idxb   = (int*)(base + off); off += (size_t)Bd * Nn * Kk;
  float* sub    = base + off; off += (size_t)Bd * Nn * Kk;
  float* Mmat   = base + off; off += (size_t)Bd * Kk * Kk;
  float* gmred  = base + off; off += (size_t)Bd * Nn * Kk;
  float* part12 = base + off; off += (size_t)NB12 * 4;
  float* st12   = base + off; off += 4;
  float* partup = base + off; off += (size_t)NBUP * 128;
  float* stup   = base + off; off += 128;
  (void)ws_size; (void)in_sizes; (void)n_in; (void)out_size;

  prep_kernel<<<(Bd * Nn + 255) / 256, 256, 0, stream>>>(x, xtp, xx);
  knn_kernel<<<Bd * (Nn / 16), 256, 0, stream>>>(xtp, xx, dist, idxb);
  sub_kernel<<<(Bd * Nn * Kk + 255) / 256, 256, 0, stream>>>(xtp, idxb, sub);
  normgram_kernel<<<Bd, 128, 0, stream>>>(sub, w_gm, Mmat);
  gmred_kernel<<<Bd * Nn * Kk / (16 * 16) / 8 * 2 / 2, 256, 0, stream>>>(
      sub, Mmat, gmred);  // 4096 tiles / 8 waves = 512 blocks
  stats12_kernel<<<NB12, 256, 0, stream>>>(dist, gmred, w_dist, part12);
  reduce12_kernel<<<1, 128, 0, stream>>>(part12, st12);
  up_kernel<1><<<NBUP, 256, 0, stream>>>(xtp, dist, idxb, gmred, st12, w_dist,
                                         g1, b1, g2, b2, w_up, g3, b3, stup,
                                         partup, out);
  reduceup_kernel<<<1, 128, 0, stream>>>(partup, stup);
  up_kernel<0><<<NBUP, 256, 0, stream>>>(xtp, dist, idxb, gmred, st12, w_dist,
                                         g1, b1, g2, b2, w_up, g3, b3, stup,
                                         partup, out);
}